// DGCNN_40097814675951
// MI455X (gfx1250) — compile-verified
//
#include <hip/hip_runtime.h>
#include <hip/hip_bf16.h>

#define BB 4
#define NN 2048
#define KK 20
#define EMBD 1024
#define CATC 512
#define BN_EPS 1e-5f

typedef __attribute__((ext_vector_type(2))) float v2f;
typedef __attribute__((ext_vector_type(8))) float v8f;

static __device__ __forceinline__ float bn_relu(float y, float g, float b) {
    float v = y * (g * rsqrtf(1.0f + BN_EPS)) + b;
    return v > 0.0f ? v : 0.0f;
}
static __device__ __forceinline__ float bn_leaky(float y, float g, float b) {
    float v = y * (g * rsqrtf(1.0f + BN_EPS)) + b;
    return v > 0.0f ? v : 0.2f * v;
}

// ---------------- transpose x [B,6,N] -> x0 [B,N,6] ----------------
__global__ void k_transpose(const float* __restrict__ x, float* __restrict__ x0) {
    int gid = blockIdx.x * blockDim.x + threadIdx.x;
    if (gid >= BB * NN * 6) return;
    int c = gid % 6;
    int row = gid / 6;
    int n = row % NN, b = row / NN;
    x0[gid] = x[((size_t)b * 6 + c) * NN + n];
}

// ---------------- squared norms per point ----------------
__global__ void k_norms(const float* __restrict__ F, int ld, int D, float* __restrict__ nrm) {
    int gid = blockIdx.x * blockDim.x + threadIdx.x;
    if (gid >= BB * NN) return;
    const float* q = F + (size_t)gid * ld;
    float s = 0.0f;
    for (int d = 0; d < D; ++d) s += q[d] * q[d];
    nrm[gid] = s;
}

// ---------------- scalar kNN for xyz (D=3, too narrow for WMMA) ----------------
__global__ void k_knn(const float* __restrict__ F, int ld, int D,
                      const float* __restrict__ nrm, int* __restrict__ idx) {
    int gid = blockIdx.x * blockDim.x + threadIdx.x;
    if (gid >= BB * NN) return;
    int b = gid / NN;
    const float* q = F + (size_t)gid * ld;
    float qn = nrm[gid];
    const float* Fb = F + (size_t)b * NN * ld;
    const float* nb = nrm + (size_t)b * NN;

    float best[KK];
    int bidx[KK];
    for (int t = 0; t < KK; ++t) { best[t] = -3.4e38f; bidx[t] = 0; }

    for (int j = 0; j < NN; ++j) {
        const float* c = Fb + (size_t)j * ld;
        __builtin_prefetch(c + ld, 0, 0);
        float inner = 0.0f;
        for (int d = 0; d < D; ++d) inner += q[d] * c[d];
        float pd = 2.0f * inner - qn - nb[j];
        if (pd > best[KK - 1]) {
            int p = KK - 1;
            while (p > 0 && best[p - 1] < pd) {
                best[p] = best[p - 1]; bidx[p] = bidx[p - 1]; --p;
            }
            best[p] = pd; bidx[p] = j;
        }
    }
    for (int t = 0; t < KK; ++t) idx[(size_t)gid * KK + t] = bidx[t];
}

// ---------------- WMMA kNN for feature layers (D multiple of 4) ----------------
// One wave owns a 16-query tile; per 16-candidate tile it computes the 16x16
// inner-product block with v_wmma_f32_16x16x4_f32, spills to LDS, and 16
// row-owner lanes run the top-20 insertion scan.
__global__ void k_knn_wmma(const float* __restrict__ F, int ld, int D,
                           const float* __restrict__ nrm, int* __restrict__ idx) {
    __shared__ float tile[8][16][17];
    int wave = threadIdx.x >> 5, lane = threadIdx.x & 31;
    int half = lane >> 4, lc = lane & 15;
    int b = blockIdx.x >> 4;                          // 16 blocks per batch
    int m0 = ((blockIdx.x & 15) * 8 + wave) << 4;     // query tile base (within batch)

    const float* Fb = F + (size_t)b * NN * ld;
    const float* nb = nrm + (size_t)b * NN;
    const float* Qr = Fb + (size_t)(m0 + lc) * ld;    // this lane's query row
    float qn = nb[m0 + lc];                           // used by lanes 0..15

    float best[KK];
    int bidx[KK];
    for (int t = 0; t < KK; ++t) { best[t] = -3.4e38f; bidx[t] = 0; }

    for (int n0 = 0; n0 < NN; n0 += 16) {
        const float* Cr = Fb + (size_t)(n0 + lc) * ld;
        v8f acc = {};
        for (int k = 0; k < D; k += 4) {
            int kk = k + 2 * half;
            v2f a;  a.x = Qr[kk];  a.y = Qr[kk + 1];
            v2f bf; bf.x = Cr[kk]; bf.y = Cr[kk + 1];
            acc = __builtin_amdgcn_wmma_f32_16x16x4_f32(false, a, false, bf,
                                                        (short)0, acc, false, false);
        }
#pragma unroll
        for (int r = 0; r < 8; ++r)
            tile[wave][r + 8 * half][lc] = acc[r];    // (row, col) of inner tile
        // wave-local LDS RAW: DS ops are in-order within a wave
        if (lane < 16) {
            for (int c = 0; c < 16; ++c) {
                float pd = 2.0f * tile[wave][lane][c] - qn - nb[n0 + c];
                if (pd > best[KK - 1]) {
                    int p = KK - 1;
                    while (p > 0 && best[p - 1] < pd) {
                        best[p] = best[p - 1]; bidx[p] = bidx[p - 1]; --p;
                    }
                    best[p] = pd; bidx[p] = n0 + c;
                }
            }
        }
    }
    if (lane < 16) {
        size_t row = (size_t)b * NN + m0 + lane;
        for (int t = 0; t < KK; ++t) idx[row * KK + t] = bidx[t];
    }
}

// ---------------- WMMA GEMM: C[M,Nc] = A[M,K] * W[Nc,K]^T ----------------
// Wave computes a 16x64 strip (4 accumulators, A fragment reused 4x). All 8
// waves of a block share one 64-row weight strip, staged into LDS with the
// CDNA5 async-to-LDS DMA path (ASYNCcnt), then read back as ds_load fragments.
__global__ void k_gemm_wmma(const float* __restrict__ A, int lda,
                            const float* __restrict__ W, int ldw,
                            float* __restrict__ C, int ldc,
                            int M, int Nc, int Kc) {
    __shared__ float sW[64 * 256];                    // up to 64 rows x Kc<=256
    int wave = threadIdx.x >> 5, lane = threadIdx.x & 31;
    int half = lane >> 4, lc = lane & 15;
    int gw = blockIdx.x * 8 + wave;
    int mtiles = M >> 4;
    int m0 = (gw % mtiles) << 4;
    int nb0 = (gw / mtiles) << 6;                     // strip base column (same for block)
    if (nb0 >= Nc) return;

    // --- async stage of weight strip: 64 rows x Kc floats ---
    int cpr = Kc >> 2;                                // 16B chunks per row
    int chunks = 64 * cpr;
    for (int t = threadIdx.x; t < chunks; t += 256) {
        int r = t / cpr, k4 = (t - r * cpr) << 2;
        unsigned lds = (unsigned)(size_t)(&sW[r * Kc + k4]);
        unsigned long long ga =
            (unsigned long long)(size_t)(W + (size_t)(nb0 + r) * ldw + k4);
        asm volatile("global_load_async_to_lds_b128 %0, %1, off"
                     :: "v"(lds), "v"(ga) : "memory");
    }
    asm volatile("s_wait_asynccnt 0x0" ::: "memory");
    __syncthreads();

    const float* Ar = A + (size_t)(m0 + lc) * lda;
    v8f acc0 = {}, acc1 = {}, acc2 = {}, acc3 = {};
    for (int k = 0; k < Kc; k += 4) {
        int kk = k + 2 * half;
        v2f a; a.x = Ar[kk]; a.y = Ar[kk + 1];
        v2f b0, b1, b2, b3;
        b0.x = sW[lc * Kc + kk];        b0.y = sW[lc * Kc + kk + 1];
        b1.x = sW[(16 + lc) * Kc + kk]; b1.y = sW[(16 + lc) * Kc + kk + 1];
        b2.x = sW[(32 + lc) * Kc + kk]; b2.y = sW[(32 + lc) * Kc + kk + 1];
        b3.x = sW[(48 + lc) * Kc + kk]; b3.y = sW[(48 + lc) * Kc + kk + 1];
        acc0 = __builtin_amdgcn_wmma_f32_16x16x4_f32(false, a, false, b0, (short)0, acc0, false, false);
        acc1 = __builtin_amdgcn_wmma_f32_16x16x4_f32(false, a, false, b1, (short)0, acc1, false, false);
        acc2 = __builtin_amdgcn_wmma_f32_16x16x4_f32(false, a, false, b2, (short)0, acc2, false, false);
        acc3 = __builtin_amdgcn_wmma_f32_16x16x4_f32(false, a, false, b3, (short)0, acc3, false, false);
    }
    int row = m0 + 8 * half;
#pragma unroll
    for (int r = 0; r < 8; ++r) {
        float* cr = C + (size_t)(row + r) * ldc + nb0 + lc;
        cr[0]  = acc0[r];
        cr[16] = acc1[r];
        cr[32] = acc2[r];
        cr[48] = acc3[r];
    }
}

// ---------------- V -= U elementwise ----------------
__global__ void k_vsub(float* __restrict__ V, const float* __restrict__ U, int n) {
    int gid = blockIdx.x * blockDim.x + threadIdx.x;
    if (gid < n) V[gid] -= U[gid];
}

// ---------------- layer0 prep: U,V,R from x0 (K = 3/9/6) ----------------
__global__ void k_l0prep(const float* __restrict__ x0, const float* __restrict__ wc,
                         const float* __restrict__ wr,
                         float* __restrict__ U, float* __restrict__ V, float* __restrict__ R) {
    int gid = blockIdx.x * blockDim.x + threadIdx.x;
    if (gid >= BB * NN * 64) return;
    int o = gid % 64, row = gid / 64;
    const float* q = x0 + (size_t)row * 6;
    const float* wcr = wc + (size_t)o * 9;
    const float* wrr = wr + (size_t)o * 6;
    float u = q[0] * wcr[0] + q[1] * wcr[1] + q[2] * wcr[2];
    float v = -u, r = 0.0f;
    for (int j = 0; j < 6; ++j) { v += q[j] * wcr[3 + j]; r += q[j] * wrr[j]; }
    U[gid] = u; V[gid] = v; R[gid] = r;
}

// ---------------- W-branch prep: Uw,Vw from xyz (ww is [C2,6]) ----------------
__global__ void k_wprep(const float* __restrict__ x0, const float* __restrict__ ww,
                        float* __restrict__ Uw, float* __restrict__ Vw, int C2) {
    int gid = blockIdx.x * blockDim.x + threadIdx.x;
    if (gid >= BB * NN * C2) return;
    int o = gid % C2, row = gid / C2;
    const float* q = x0 + (size_t)row * 6;
    const float* w = ww + (size_t)o * 6;
    float u = q[0] * w[0] + q[1] * w[1] + q[2] * w[2];
    float v = q[0] * w[3] + q[1] * w[4] + q[2] * w[5] - u;
    Uw[gid] = u; Vw[gid] = v;
}

// ---------------- gather-max + BN + ReLU + W-modulation + residual ----------------
__global__ void k_edgemax(const float* __restrict__ U, const float* __restrict__ V,
                          const float* __restrict__ Uw, const float* __restrict__ Vw,
                          const float* __restrict__ R,
                          const int* __restrict__ idxc, const int* __restrict__ idxw,
                          const float* __restrict__ gc, const float* __restrict__ bc,
                          const float* __restrict__ gw, const float* __restrict__ bw,
                          float* __restrict__ out, int C2) {
    int gid = blockIdx.x * blockDim.x + threadIdx.x;
    if (gid >= BB * NN * C2) return;
    int o = gid % C2, row = gid / C2;
    size_t bbase = (size_t)(row / NN) * NN;
    const int* ic = idxc + (size_t)row * KK;
    const int* iw = idxw + (size_t)row * KK;
    float mc = -3.4e38f, mw = -3.4e38f;
    for (int k = 0; k < KK; ++k) {
        mc = fmaxf(mc, U[(bbase + ic[k]) * C2 + o]);
        mw = fmaxf(mw, Uw[(bbase + iw[k]) * C2 + o]);
    }
    float xc = bn_relu(mc + V[(size_t)row * C2 + o], gc[o], bc[o]);
    float xw = bn_relu(mw + Vw[(size_t)row * C2 + o], gw[o], bw[o]);
    out[(size_t)row * CATC + o] = xc * xw + R[(size_t)row * C2 + o];
}

// ---------------- fused pre-pool GEMM (WMMA) + BN + leaky + max/mean over N ----------------
__global__ void k_prepool(const float* __restrict__ cat, const float* __restrict__ wpre,
                          const float* __restrict__ gpre, const float* __restrict__ bpre,
                          float* __restrict__ hmax, float* __restrict__ hmean) {
    __shared__ float smax[256], ssum[256];
    int b = blockIdx.x >> 6;
    int e0 = (blockIdx.x & 63) << 4;
    int wave = threadIdx.x >> 5, lane = threadIdx.x & 31;
    int half = lane >> 4, lc = lane & 15;
    float sc = gpre[e0 + lc] * rsqrtf(1.0f + BN_EPS);
    float bi = bpre[e0 + lc];
    const float* Wr = wpre + (size_t)(e0 + lc) * CATC;
    float lmax = -3.4e38f, ls = 0.0f;

    for (int jt = wave; jt < (NN / 16); jt += 8) {
        int m0 = jt << 4;
        const float* Ar = cat + ((size_t)b * NN + m0 + lc) * CATC;
        v8f acc = {};
        for (int k = 0; k < CATC; k += 4) {
            int kk = k + 2 * half;
            v2f a; a.x = Ar[kk]; a.y = Ar[kk + 1];
            v2f w; w.x = Wr[kk]; w.y = Wr[kk + 1];
            acc = __builtin_amdgcn_wmma_f32_16x16x4_f32(false, a, false, w,
                                                        (short)0, acc, false, false);
        }
#pragma unroll
        for (int r = 0; r < 8; ++r) {
            float v = acc[r] * sc + bi;
            v = v > 0.0f ? v : 0.2f * v;
            lmax = fmaxf(lmax, v);
            ls += v;
        }
    }
    smax[threadIdx.x] = lmax;
    ssum[threadIdx.x] = ls;
    __syncthreads();
    if (threadIdx.x < 16) {
        float m = -3.4e38f, s = 0.0f;
        for (int t = threadIdx.x; t < 256; t += 16) { m = fmaxf(m, smax[t]); s += ssum[t]; }
        hmax[(size_t)b * EMBD + e0 + threadIdx.x] = m;
        hmean[(size_t)b * EMBD + e0 + threadIdx.x] = s * (1.0f / NN);
    }
}

// ---------------- final FC layers (tiny) ----------------
__global__ void k_fc1(const float* __restrict__ hmax, const float* __restrict__ hmean,
                      const float* __restrict__ w1, const float* __restrict__ g1,
                      const float* __restrict__ bb1, float* __restrict__ h1) {
    int gid = blockIdx.x * blockDim.x + threadIdx.x;
    if (gid >= BB * 512) return;
    int b = gid / 512, o = gid % 512;
    const float* wr = w1 + (size_t)o * 2048;
    float s = 0.0f;
    for (int e = 0; e < EMBD; ++e) s += hmax[(size_t)b * EMBD + e] * wr[e];
    for (int e = 0; e < EMBD; ++e) s += hmean[(size_t)b * EMBD + e] * wr[EMBD + e];
    h1[gid] = bn_leaky(s, g1[o], bb1[o]);
}

__global__ void k_fc2(const float* __restrict__ h1, const float* __restrict__ w2,
                      const float* __restrict__ b2, const float* __restrict__ g2,
                      const float* __restrict__ bb2, float* __restrict__ h2) {
    int gid = blockIdx.x * blockDim.x + threadIdx.x;
    if (gid >= BB * 256) return;
    int b = gid / 256, o = gid % 256;
    const float* wr = w2 + (size_t)o * 512;
    float s = b2[o];
    for (int e = 0; e < 512; ++e) s += h1[(size_t)b * 512 + e] * wr[e];
    h2[gid] = bn_leaky(s, g2[o], bb2[o]);
}

__global__ void k_fc3(const float* __restrict__ h2, const float* __restrict__ w3,
                      const float* __restrict__ b3, float* __restrict__ out) {
    int gid = blockIdx.x * blockDim.x + threadIdx.x;
    if (gid >= BB * 40) return;
    int b = gid / 40, o = gid % 40;
    const float* wr = w3 + (size_t)o * 256;
    float s = b3[o];
    for (int e = 0; e < 256; ++e) s += h2[(size_t)b * 256 + e] * wr[e];
    out[gid] = s;
}

// ---------------- host side ----------------
struct ParamPtrs {
    const float* x;
    const float *wc[4], *gc[4], *bc[4], *ww[4], *gw[4], *bw[4], *wr[4];
    const float *wpre, *gpre, *bpre, *w1, *g1, *bb1, *w2, *b2, *g2, *bb2, *w3, *b3;
};

static inline int cdiv(int a, int b) { return (a + b - 1) / b; }

extern "C" void kernel_launch(void* const* d_in, const int* in_sizes, int n_in,
                              void* d_out, int out_size, void* d_ws, size_t ws_size,
                              hipStream_t stream) {
    (void)out_size; (void)ws_size;
    ParamPtrs p;
    int i = 0;
    auto F = [&](int k) { return (const float*)d_in[k]; };
    if (n_in >= 1 && in_sizes[0] == BB * 6 * NN) {
        // recursive insertion order: x, then params
        p.x = F(i++);
        for (int l = 0; l < 4; ++l) {
            p.wc[l] = F(i++); p.gc[l] = F(i++); p.bc[l] = F(i++);
            p.ww[l] = F(i++); p.gw[l] = F(i++); p.bw[l] = F(i++);
            p.wr[l] = F(i++);
        }
        p.wpre = F(i++); p.gpre = F(i++); p.bpre = F(i++);
        p.w1 = F(i++); p.g1 = F(i++); p.bb1 = F(i++);
        p.w2 = F(i++); p.b2 = F(i++); p.g2 = F(i++); p.bb2 = F(i++);
        p.w3 = F(i++); p.b3 = F(i++);
    } else {
        // jax pytree (alphabetical) order: params leaves sorted, then x
        p.b2 = F(i++); p.b3 = F(i++); p.bb1 = F(i++); p.bb2 = F(i++); p.bpre = F(i++);
        p.g1 = F(i++); p.g2 = F(i++); p.gpre = F(i++);
        for (int l = 0; l < 4; ++l) {
            p.bc[l] = F(i++); p.bw[l] = F(i++); p.gc[l] = F(i++); p.gw[l] = F(i++);
            p.wc[l] = F(i++); p.wr[l] = F(i++); p.ww[l] = F(i++);
        }
        p.w1 = F(i++); p.w2 = F(i++); p.w3 = F(i++); p.wpre = F(i++);
        p.x = F(i++);
    }

    // workspace layout (float offsets)
    float* ws = (float*)d_ws;
    float* x0    = ws;                        // 49152
    float* nrm   = ws + 49152;                // 8192
    int*   idx0  = (int*)(ws + 57344);        // 163840
    int*   idxl  = (int*)(ws + 221184);       // 163840
    float* U     = ws + 385024;               // 2097152
    float* V     = ws + 2482176;              // 2097152
    float* Uw    = ws + 4579328;              // 2097152
    float* Vw    = ws + 6676480;              // 2097152
    float* R     = ws + 8773632;              // 2097152
    float* cat   = ws + 10870784;             // 4194304  [B,N,512]
    float* hmax  = ws + 15065088;             // 4096
    float* hmean = ws + 15069184;             // 4096
    float* h1    = ws + 15073280;             // 2048
    float* h2    = ws + 15075328;             // 1024

    const int C2arr[4] = {64, 64, 128, 256};
    const int offarr[4] = {0, 64, 128, 256};
    const int M = BB * NN;                    // 8192

    k_transpose<<<cdiv(BB * NN * 6, 256), 256, 0, stream>>>(p.x, x0);

    // shared xyz kNN (used by layer0 conv-branch and all W-branches)
    k_norms<<<cdiv(M, 256), 256, 0, stream>>>(x0, 6, 3, nrm);
    k_knn<<<cdiv(M, 256), 256, 0, stream>>>(x0, 6, 3, nrm, idx0);

    // ---- layer 0 ----
    k_l0prep<<<cdiv(M * 64, 256), 256, 0, stream>>>(x0, p.wc[0], p.wr[0], U, V, R);
    k_wprep<<<cdiv(M * 64, 256), 256, 0, stream>>>(x0, p.ww[0], Uw, Vw, 64);
    k_edgemax<<<cdiv(M * 64, 256), 256, 0, stream>>>(U, V, Uw, Vw, R, idx0, idx0,
                                                     p.gc[0], p.bc[0], p.gw[0], p.bw[0],
                                                     cat + offarr[0], 64);

    // ---- layers 1..3 (WMMA kNN + WMMA GEMMs) ----
    for (int l = 1; l < 4; ++l) {
        int Cp = C2arr[l - 1], C2 = C2arr[l];
        const float* Fl = cat + offarr[l - 1];
        k_norms<<<cdiv(M, 256), 256, 0, stream>>>(Fl, CATC, Cp, nrm);
        k_knn_wmma<<<BB * 16, 256, 0, stream>>>(Fl, CATC, Cp, nrm, idxl);

        int blocks = (M / 16) * (C2 / 64) / 8;    // waves = mtiles * nstrips
        k_gemm_wmma<<<blocks, 256, 0, stream>>>(Fl, CATC, p.wc[l], 2 * Cp,
                                                U, C2, M, C2, Cp);
        k_gemm_wmma<<<blocks, 256, 0, stream>>>(Fl, CATC, p.wc[l] + Cp, 2 * Cp,
                                                V, C2, M, C2, Cp);
        k_vsub<<<cdiv(M * C2, 256), 256, 0, stream>>>(V, U, M * C2);
        k_gemm_wmma<<<blocks, 256, 0, stream>>>(Fl, CATC, p.wr[l], Cp,
                                                R, C2, M, C2, Cp);
        k_wprep<<<cdiv(M * C2, 256), 256, 0, stream>>>(x0, p.ww[l], Uw, Vw, C2);
        k_edgemax<<<cdiv(M * C2, 256), 256, 0, stream>>>(U, V, Uw, Vw, R, idxl, idx0,
                                                         p.gc[l], p.bc[l], p.gw[l], p.bw[l],
                                                         cat + offarr[l], C2);
    }

    // ---- fused pre-pool GEMM + pooling ----
    k_prepool<<<BB * 64, 256, 0, stream>>>(cat, p.wpre, p.gpre, p.bpre, hmax, hmean);

    // ---- classifier head ----
    k_fc1<<<cdiv(BB * 512, 256), 256, 0, stream>>>(hmax, hmean, p.w1, p.g1, p.bb1, h1);
    k_fc2<<<cdiv(BB * 256, 256), 256, 0, stream>>>(h1, p.w2, p.b2, p.g2, p.bb2, h2);
    k_fc3<<<cdiv(BB * 40, 128), 128, 0, stream>>>(h2, p.w3, p.b3, (float*)d_out);
}